// NodeReduceModule_67705864454153
// MI455X (gfx1250) — compile-verified
//
#include <hip/hip_runtime.h>
#include <math.h>

// Problem constants (match reference)
#define NN    32768
#define DIN   512
#define DHID  512
#define HH    4
#define LEAKY 0.01f

typedef __attribute__((ext_vector_type(2))) float v2f;
typedef __attribute__((ext_vector_type(8))) float v8f;

// ---------------------------------------------------------------------------
// Phase 1: pv[h] = fc_w[h]^T wv[h], pu[h] = fc_w[h]^T wu[h]
// Stored interleaved as Pc[k][h], k in [0,1024): k<512 -> pv (hv part),
// k>=512 -> pu (hu part).  4 MB read, trivial.
// ---------------------------------------------------------------------------
__global__ void __launch_bounds__(256)
k_proj(const float* __restrict__ fc_w, const float* __restrict__ att_w,
       float* __restrict__ Pc) {
  int t = blockIdx.x * 256 + threadIdx.x;   // 0..2047
  int h = t >> 9;
  int i = t & (DIN - 1);
  const float* W  = fc_w + (size_t)h * DHID * DIN;
  const float* av = att_w + h * 2 * DHID;     // wv
  const float* au = av + DHID;                // wu
  float pv = 0.f, pu = 0.f;
  for (int o = 0; o < DHID; ++o) {
    float w = W[(size_t)o * DIN + i];         // coalesced across i
    pv = fmaf(w, av[o], pv);
    pu = fmaf(w, au[o], pu);
  }
  Pc[i * HH + h]          = pv;
  Pc[(DIN + i) * HH + h]  = pu;
}

// ---------------------------------------------------------------------------
// Phase 2: logits[h,n] = hv[n]·pv[h] + hu[n]·pu[h], via V_WMMA_F32_16X16X4_F32.
// Wave = one 16-row M tile, K=1024 in steps of 4. A fragment: lane L holds
// A[L&15][kb+2*(L>>4)+{0,1}] -> one b64 load/lane/step (fully utilized).
// B fragment from LDS copy of Pc; only N columns 0..3 are meaningful, the
// rest compute garbage that is never stored. Leaky-ReLU fused in epilogue.
// ---------------------------------------------------------------------------
__global__ void __launch_bounds__(256)
k_logits(const float* __restrict__ hv, const float* __restrict__ hu,
         const float* __restrict__ Pc, float* __restrict__ logits) {
  __shared__ float Pl[1024 * HH];           // 16 KB
  for (int idx = threadIdx.x; idx < 1024 * HH; idx += 256) Pl[idx] = Pc[idx];
  __syncthreads();

  const int lane    = threadIdx.x & 31;
  const int wave    = threadIdx.x >> 5;
  const int half    = lane >> 4;            // 0 or 1
  const int rowin   = lane & 15;
  const int nm      = lane & 3;             // head column (replicated for n>=4)
  const int rowbase = blockIdx.x * 128 + wave * 16;

  const float* rv = hv + (size_t)(rowbase + rowin) * DIN;
  const float* ru = hu + (size_t)(rowbase + rowin) * DIN;

  v8f acc = {};
  // hv half of K (Pv columns)
  for (int kb = 0; kb < DIN; kb += 4) {
    int k0 = kb + 2 * half;
    v2f a = *(const v2f*)(rv + k0);                 // 8B aligned
    v2f b;
    b.x = Pl[k0 * HH + nm];
    b.y = Pl[(k0 + 1) * HH + nm];
    acc = __builtin_amdgcn_wmma_f32_16x16x4_f32(false, a, false, b,
                                                (short)0, acc, false, false);
  }
  // hu half of K (Pu columns)
  for (int kb = 0; kb < DIN; kb += 4) {
    int k0 = kb + 2 * half;
    v2f a = *(const v2f*)(ru + k0);
    v2f b;
    b.x = Pl[(DIN + k0) * HH + nm];
    b.y = Pl[(DIN + k0 + 1) * HH + nm];
    acc = __builtin_amdgcn_wmma_f32_16x16x4_f32(false, a, false, b,
                                                (short)0, acc, false, false);
  }

  // D layout: VGPR j, lane L -> row = j + 8*(L>>4), col = L&15.
  if ((lane & 15) < HH) {
    int h = lane & 15;
#pragma unroll
    for (int j = 0; j < 8; ++j) {
      int row = rowbase + j + 8 * half;
      float l = acc[j];
      l = (l >= 0.f) ? l : LEAKY * l;               // fused leaky-ReLU
      logits[(size_t)h * NN + row] = l;
    }
  }
}

// ---------------------------------------------------------------------------
// Phase 3: per-head softmax stats (max, sum exp) over N=32768.
// One 1024-thread block per head, fixed-order tree reduction (deterministic).
// ---------------------------------------------------------------------------
__global__ void __launch_bounds__(1024)
k_softmax_stats(const float* __restrict__ logits, float* __restrict__ stats) {
  __shared__ float red[1024];
  const int h = blockIdx.x;
  const float* L = logits + (size_t)h * NN;
  const int t = threadIdx.x;

  float m = -__builtin_inff();
  for (int i = t; i < NN; i += 1024) m = fmaxf(m, L[i]);
  red[t] = m; __syncthreads();
  for (int s = 512; s > 0; s >>= 1) {
    if (t < s) red[t] = fmaxf(red[t], red[t + s]);
    __syncthreads();
  }
  float mx = red[0];
  __syncthreads();

  float z = 0.f;
  for (int i = t; i < NN; i += 1024) z += __expf(L[i] - mx);
  red[t] = z; __syncthreads();
  for (int s = 512; s > 0; s >>= 1) {
    if (t < s) red[t] += red[t + s];
    __syncthreads();
  }
  if (t == 0) { stats[h * 2] = mx; stats[h * 2 + 1] = red[0]; }
}

// ---------------------------------------------------------------------------
// Phase 4a: block-local weighted column sums of hv (all 4 heads share one
// pass over hv).  Block b owns rows [256b, 256b+256); partials to ws.
// ---------------------------------------------------------------------------
__global__ void __launch_bounds__(256)
k_wsum_part(const float* __restrict__ hv, const float* __restrict__ logits,
            const float* __restrict__ stats, float* __restrict__ part) {
  __shared__ float w[HH][256];
  const int t = threadIdx.x;
  const int rowbase = blockIdx.x * 256;
#pragma unroll
  for (int h = 0; h < HH; ++h)
    w[h][t] = __expf(logits[(size_t)h * NN + rowbase + t] - stats[h * 2]);
  __syncthreads();

  float a00=0,a01=0,a10=0,a11=0,a20=0,a21=0,a30=0,a31=0;
  const float* base = hv + (size_t)rowbase * DIN;
  for (int n = 0; n < 256; ++n) {
    float x0 = base[(size_t)n * DIN + t];         // coalesced
    float x1 = base[(size_t)n * DIN + t + 256];   // coalesced
    float w0 = w[0][n], w1 = w[1][n], w2 = w[2][n], w3 = w[3][n]; // broadcast
    a00 = fmaf(w0, x0, a00); a01 = fmaf(w0, x1, a01);
    a10 = fmaf(w1, x0, a10); a11 = fmaf(w1, x1, a11);
    a20 = fmaf(w2, x0, a20); a21 = fmaf(w2, x1, a21);
    a30 = fmaf(w3, x0, a30); a31 = fmaf(w3, x1, a31);
  }
  float* P = part + (size_t)blockIdx.x * HH * DIN;
  P[0 * DIN + t] = a00; P[0 * DIN + t + 256] = a01;
  P[1 * DIN + t] = a10; P[1 * DIN + t + 256] = a11;
  P[2 * DIN + t] = a20; P[2 * DIN + t + 256] = a21;
  P[3 * DIN + t] = a30; P[3 * DIN + t + 256] = a31;
}

// Phase 4b: deterministic reduce of the 128 partials; normalize by Z.
__global__ void __launch_bounds__(256)
k_wsum_reduce(const float* __restrict__ part, const float* __restrict__ stats,
              float* __restrict__ svec) {
  int t = blockIdx.x * 256 + threadIdx.x;   // 0..2047
  int h = t >> 9;
  int i = t & (DIN - 1);
  float s = 0.f;
  for (int b = 0; b < 128; ++b)
    s += part[(size_t)b * HH * DIN + h * DIN + i];
  svec[h * DIN + i] = s / stats[h * 2 + 1];
}

// ---------------------------------------------------------------------------
// Phase 5: out[h,o] = fc_w[h][o,:] · s[h].  One wave per output element,
// lanes stride the 512-wide dot, wave32 xor-shuffle reduction.
// ---------------------------------------------------------------------------
__global__ void __launch_bounds__(256)
k_out(const float* __restrict__ fc_w, const float* __restrict__ svec,
      float* __restrict__ out) {
  const int wave = threadIdx.x >> 5;
  const int lane = threadIdx.x & 31;
  const int idx  = blockIdx.x * 8 + wave;   // 0..63
  const int h = idx >> 9;
  const int o = idx & (DHID - 1);
  const float* W = fc_w + (size_t)(h * DHID + o) * DIN;
  const float* s = svec + h * DIN;
  float acc = 0.f;
#pragma unroll 4
  for (int j = 0; j < 16; ++j) {
    int i = lane + 32 * j;
    acc = fmaf(W[i], s[i], acc);
  }
#pragma unroll
  for (int off = 16; off > 0; off >>= 1)
    acc += __shfl_xor(acc, off, 32);
  if (lane == 0) out[idx] = acc;
}

// ---------------------------------------------------------------------------
extern "C" void kernel_launch(void* const* d_in, const int* in_sizes, int n_in,
                              void* d_out, int out_size, void* d_ws, size_t ws_size,
                              hipStream_t stream) {
  const float* hu    = (const float*)d_in[0];
  const float* hv    = (const float*)d_in[1];
  const float* fc_w  = (const float*)d_in[2];
  const float* att_w = (const float*)d_in[3];
  float* out = (float*)d_out;

  // Workspace layout (bytes), ~1.6 MB total
  char* ws = (char*)d_ws;
  float* Pc     = (float*)(ws + 0);                       // 1024*4 f32 = 16 KB
  float* logits = (float*)(ws + 16384);                   // 4*32768 f32 = 512 KB
  float* stats  = (float*)(ws + 16384 + 524288);          // 8 f32
  float* part   = (float*)(ws + 16384 + 524288 + 4096);   // 128*4*512 f32 = 1 MB
  float* svec   = (float*)(ws + 16384 + 524288 + 4096 + 1048576); // 2048 f32

  k_proj<<<(HH * DIN) / 256, 256, 0, stream>>>(fc_w, att_w, Pc);
  k_logits<<<NN / 128, 256, 0, stream>>>(hv, hu, Pc, logits);
  k_softmax_stats<<<HH, 1024, 0, stream>>>(logits, stats);
  k_wsum_part<<<NN / 256, 256, 0, stream>>>(hv, logits, stats, part);
  k_wsum_reduce<<<(HH * DIN) / 256, 256, 0, stream>>>(part, stats, svec);
  k_out<<<(HH * DHID) / 8 / 32, 256, 0, stream>>>(fc_w, svec, out);
}